// TripletLossCloth_12223476924977
// MI455X (gfx1250) — compile-verified
//
#include <hip/hip_runtime.h>
#include <hip/hip_bf16.h>

typedef float v2f __attribute__((ext_vector_type(2)));
typedef float v8f __attribute__((ext_vector_type(8)));

#define MARGIN 0.3f
#define BIGF   1e30f

// ---------------------------------------------------------------------------
// ws layout: [0..16)   double sums[2]  (term_sum, count)
//            [64..)    float   sq[n]
//            then      uint    an_min[n]   (float bits, min via uint order)
//            then      uint    ap2[n]      (float bits, max via uint order)
//            then      int     any1[n]
// ---------------------------------------------------------------------------

__global__ __launch_bounds__(256)
void tl_init_kernel(unsigned* __restrict__ an, unsigned* __restrict__ ap,
                    int* __restrict__ any1, double* __restrict__ sums, int n) {
    int i = blockIdx.x * blockDim.x + threadIdx.x;
    if (i < n) {
        an[i] = __float_as_uint(BIGF);
        ap[i] = 0u;              // dist >= 0 always (diag is cm==2), so 0.0f is a safe -inf
        any1[i] = 0;
    }
    if (i == 0) { sums[0] = 0.0; sums[1] = 0.0; }
}

// one wave per row: sq[i] = sum_k X[i][k]^2
__global__ __launch_bounds__(256)
void tl_sq_kernel(const float* __restrict__ X, float* __restrict__ sq, int n, int d) {
    int wave = threadIdx.x >> 5, lane = threadIdx.x & 31;
    int i = blockIdx.x * 8 + wave;
    if (i >= n) return;
    const float4* xi = (const float4*)(X + (size_t)i * d);
    float s = 0.f;
    for (int c = lane; c < (d >> 2); c += 32) {
        float4 a = xi[c];
        s += a.x * a.x + a.y * a.y + a.z * a.z + a.w * a.w;
    }
    for (int off = 16; off >= 1; off >>= 1) s += __shfl_xor(s, off, 32);
    if (lane == 0) sq[i] = s;
}

// ---------------------------------------------------------------------------
// Fused symmetric Gram pass.  Block = 256 thr = 8 waves, one 128x128 tile
// with Jb >= Ib (triangular sweep -> half the GEMM).  Each wave owns a
// 16x128 strip (8 v8f accumulators).  The tile is reduced in BOTH
// orientations: row-side stats for rows in Ib, column-side stats (via the
// symmetry dist[i,j]==dist[j,i]) for rows in Jb.  All stats are idempotent
// (min/max/or), so diagonal-tile double coverage is harmless.
// ---------------------------------------------------------------------------
#define KC      32
#define LSTRIDE 36   // even (keeps float2 8B-aligned), 16 rows hit 16 distinct banks

__global__ __launch_bounds__(256)
void tl_gram_stats_kernel(const float* __restrict__ X,
                          const int* __restrict__ tgt, const int* __restrict__ cid,
                          const float* __restrict__ sq,
                          unsigned* __restrict__ an_min, unsigned* __restrict__ ap2,
                          int* __restrict__ any1, int n, int d) {
    const int Ib = blockIdx.x, Jb = blockIdx.y;
    if (Jb < Ib) return;                      // triangular sweep

    __shared__ float la[128 * LSTRIDE];
    __shared__ float lb[128 * LSTRIDE];
    __shared__ unsigned s_cmn[128], s_cmx[128];
    __shared__ int s_cany[128];

    const int tid   = threadIdx.x;
    const int wave  = tid >> 5;
    const int lane  = tid & 31;
    const int hi    = lane >> 4;      // which half of the wave
    const int lm    = lane & 15;
    const int wrow  = wave * 16;      // wave's row offset inside the 128-row block
    const int ibase = Ib * 128;
    const int jbase = Jb * 128;

    if (tid < 128) {
        s_cmn[tid]  = __float_as_uint(BIGF);
        s_cmx[tid]  = 0u;
        s_cany[tid] = 0;
    }

    // per-lane row metadata (rows v + 8*hi of this wave's strip)
    int   ti8[8], ci8[8];
    float sqi8[8];
#pragma unroll
    for (int v = 0; v < 8; ++v) {
        int i   = ibase + wrow + v + 8 * hi;
        ti8[v]  = tgt[i];
        ci8[v]  = cid[i];
        sqi8[v] = sq[i];
    }

    v8f acc[8];
#pragma unroll
    for (int t = 0; t < 8; ++t)
#pragma unroll
        for (int e = 0; e < 8; ++e) acc[t][e] = 0.f;

    for (int kc = 0; kc < d; kc += KC) {
        // stage 128x32 A block (rows Ib) and 128x32 B block (rows Jb) into LDS
#pragma unroll
        for (int it = 0; it < 4; ++it) {
            int idx = tid + it * 256;          // 1024 float4 per matrix per block
            int r   = idx >> 3;
            int c4  = (idx & 7) * 4;
            float4 va = *(const float4*)(X + (size_t)(ibase + r) * d + kc + c4);
            *(float4*)(&la[r * LSTRIDE + c4]) = va;
            float4 vb = *(const float4*)(X + (size_t)(jbase + r) * d + kc + c4);
            *(float4*)(&lb[r * LSTRIDE + c4]) = vb;
        }
        __syncthreads();

#pragma unroll
        for (int ks = 0; ks < KC; ks += 4) {
            int k0 = ks + 2 * hi;              // lane's K pair within this 4-step
            v2f a = *(const v2f*)(&la[(wrow + lm) * LSTRIDE + k0]);
#pragma unroll
            for (int tn = 0; tn < 8; ++tn) {
                v2f b = *(const v2f*)(&lb[(tn * 16 + lm) * LSTRIDE + k0]);
                acc[tn] = __builtin_amdgcn_wmma_f32_16x16x4_f32(
                    false, a, false, b, (short)0, acc[tn], false, false);
            }
        }
        __syncthreads();
    }

    // ---- epilogue: Gram -> dist, masks, row-side AND column-side stats ----
    float rmin[8], rmax2[8];           // per v  (rows of Ib)
    int   rany[8];
    float cmin[8], cmax2[8];           // per tn (columns == rows of Jb)
    int   cany[8];
#pragma unroll
    for (int v = 0; v < 8; ++v) {
        rmin[v] = BIGF; rmax2[v] = 0.f; rany[v] = 0;
        cmin[v] = BIGF; cmax2[v] = 0.f; cany[v] = 0;
    }

#pragma unroll
    for (int tn = 0; tn < 8; ++tn) {
        int   j   = jbase + tn * 16 + lm;      // this lane's column
        int   tj  = tgt[j], cj = cid[j];
        float sqj = sq[j];
#pragma unroll
        for (int v = 0; v < 8; ++v) {
            float s    = acc[tn][v];
            float dist = sqrtf(fmaxf(sqi8[v] + sqj - 2.f * s, 1e-12f));
            bool  sid  = (ti8[v] == tj);
            bool  scl  = (ci8[v] == cj);
            if (!sid) {
                rmin[v]  = fminf(rmin[v],  dist);   // hardest negative (row)
                cmin[tn] = fminf(cmin[tn], dist);   // symmetric: row j, column i
            }
            if (sid && scl) {
                rmax2[v]  = fmaxf(rmax2[v],  dist); // cm == 2
                cmax2[tn] = fmaxf(cmax2[tn], dist);
            }
            if (sid != scl) { rany[v] = 1; cany[tn] = 1; }   // cm == 1
        }
    }

    // row-side: reduce across the 16 lanes sharing each row (xor stays in-half)
#pragma unroll
    for (int v = 0; v < 8; ++v) {
        float mn = rmin[v], mx = rmax2[v];
        int   an = rany[v];
        for (int off = 8; off >= 1; off >>= 1) {
            mn  = fminf(mn, __shfl_xor(mn, off, 32));
            mx  = fmaxf(mx, __shfl_xor(mx, off, 32));
            an |= __shfl_xor(an, off, 32);
        }
        if (lm == 0) {
            int i = ibase + wrow + v + 8 * hi;
            atomicMin(&an_min[i], __float_as_uint(mn));   // valid: all values >= 0
            atomicMax(&ap2[i],    __float_as_uint(mx));
            if (an) atomicOr(&any1[i], 1);
        }
    }

    // column-side: a column's 16 elements live in lanes l and l^16
    __syncthreads();   // s_c* init visible
#pragma unroll
    for (int tn = 0; tn < 8; ++tn) {
        float mn = fminf(cmin[tn],  __shfl_xor(cmin[tn],  16, 32));
        float mx = fmaxf(cmax2[tn], __shfl_xor(cmax2[tn], 16, 32));
        int   an = cany[tn] | __shfl_xor(cany[tn], 16, 32);
        if (hi == 0) {                          // lanes 0-15 publish per wave
            int c = tn * 16 + lm;
            atomicMin(&s_cmn[c], __float_as_uint(mn));
            atomicMax(&s_cmx[c], __float_as_uint(mx));
            if (an) atomicOr(&s_cany[c], 1);
        }
    }
    __syncthreads();
    if (tid < 128) {
        int j = jbase + tid;
        atomicMin(&an_min[j], s_cmn[tid]);
        atomicMax(&ap2[j],    s_cmx[tid]);
        if (s_cany[tid]) atomicOr(&any1[j], 1);
    }
}

// term1: rows with no cm==1 entry contribute relu(ap2 - an_min + margin)
__global__ __launch_bounds__(256)
void tl_term1_kernel(const unsigned* __restrict__ an, const unsigned* __restrict__ ap,
                     const int* __restrict__ any1, double* __restrict__ sums, int n) {
    __shared__ double s0[256], s1[256];
    int tid = threadIdx.x;
    int i   = blockIdx.x * blockDim.x + tid;
    double t = 0.0, c = 0.0;
    if (i < n && !any1[i]) {
        float v = __uint_as_float(ap[i]) - __uint_as_float(an[i]) + MARGIN;
        t = (v > 0.f) ? (double)v : 0.0;
        c = 1.0;
    }
    s0[tid] = t; s1[tid] = c;
    __syncthreads();
    for (int s = 128; s > 0; s >>= 1) {
        if (tid < s) { s0[tid] += s0[tid + s]; s1[tid] += s1[tid + s]; }
        __syncthreads();
    }
    if (tid == 0) { atomicAdd(&sums[0], s0[0]); atomicAdd(&sums[1], s1[0]); }
}

// term2: sparse cm==1 pairs; one wave per row sweeps columns, recomputes the
// distance only on hits (~N/1000 + N/3000 per row, all L2-resident)
__global__ __launch_bounds__(256)
void tl_term2_kernel(const float* __restrict__ X,
                     const int* __restrict__ tgt, const int* __restrict__ cid,
                     const float* __restrict__ sq, const unsigned* __restrict__ an,
                     double* __restrict__ sums, int n, int d) {
    __shared__ double s0[256], s1[256];
    int tid  = threadIdx.x;
    int wave = tid >> 5, lane = tid & 31;
    int i    = blockIdx.x * 8 + wave;

    double lsum = 0.0, lcnt = 0.0;
    if (i < n) {
        int   ti  = tgt[i], ci = cid[i];
        float sqi = sq[i];
        float ani = __uint_as_float(an[i]);
        const float4* xi = (const float4*)(X + (size_t)i * d);
        for (int j = lane; j < n; j += 32) {
            bool one = (tgt[j] == ti) != (cid[j] == ci);   // cm == 1
            if (one) {
                const float4* xj = (const float4*)(X + (size_t)j * d);
                float dot = 0.f;
                for (int c = 0; c < (d >> 2); ++c) {
                    float4 a = xi[c], b = xj[c];
                    dot += a.x * b.x + a.y * b.y + a.z * b.z + a.w * b.w;
                }
                float dist = sqrtf(fmaxf(sqi + sq[j] - 2.f * dot, 1e-12f));
                float v = dist - ani + MARGIN;
                if (v > 0.f) lsum += (double)v;
                lcnt += 1.0;
            }
        }
    }
    s0[tid] = lsum; s1[tid] = lcnt;
    __syncthreads();
    for (int s = 128; s > 0; s >>= 1) {
        if (tid < s) { s0[tid] += s0[tid + s]; s1[tid] += s1[tid + s]; }
        __syncthreads();
    }
    if (tid == 0) { atomicAdd(&sums[0], s0[0]); atomicAdd(&sums[1], s1[0]); }
}

__global__ void tl_finalize_kernel(const double* __restrict__ sums, float* __restrict__ out) {
    double c = sums[1];
    if (c < 1.0) c = 1.0;
    out[0] = (float)(sums[0] / c);
}

extern "C" void kernel_launch(void* const* d_in, const int* in_sizes, int n_in,
                              void* d_out, int out_size, void* d_ws, size_t ws_size,
                              hipStream_t stream) {
    const float* X   = (const float*)d_in[0];
    const int*   tgt = (const int*)d_in[1];
    const int*   cid = (const int*)d_in[2];

    const int n = in_sizes[1];           // 8192
    const int d = in_sizes[0] / n;       // 512

    char*     ws   = (char*)d_ws;
    double*   sums = (double*)ws;                         // 2 doubles
    float*    sq   = (float*)(ws + 64);
    unsigned* an   = (unsigned*)(ws + 64 + (size_t)n * 4);
    unsigned* ap   = (unsigned*)(ws + 64 + (size_t)n * 8);
    int*      any1 = (int*)(ws + 64 + (size_t)n * 12);

    tl_init_kernel<<<(n + 255) / 256, 256, 0, stream>>>(an, ap, any1, sums, n);
    tl_sq_kernel<<<n / 8, 256, 0, stream>>>(X, sq, n, d);

    dim3 grid(n / 128, n / 128);         // triangular: blocks with Jb < Ib exit
    tl_gram_stats_kernel<<<grid, 256, 0, stream>>>(X, tgt, cid, sq, an, ap, any1, n, d);

    tl_term1_kernel<<<(n + 255) / 256, 256, 0, stream>>>(an, ap, any1, sums, n);
    tl_term2_kernel<<<n / 8, 256, 0, stream>>>(X, tgt, cid, sq, an, sums, n, d);
    tl_finalize_kernel<<<1, 1, 0, stream>>>(sums, (float*)d_out);
}